// FastFoodLayer_53807350284471
// MI455X (gfx1250) — compile-verified
//
#include <hip/hip_runtime.h>

// FastFood layer for MI455X (gfx1250): FWHT_4096 = (H16 (x) H16 (x) H16) done as
// three batched 16x16x16 GEMM passes per row using V_WMMA_F32_16X16X4_F32.
// One wave32 owns one row staged in LDS; no barriers (DS ops are in-order per wave).
// Tile loops are capped at unroll-4 to keep VGPR usage < 256 (avoids
// s_set_vgpr_msb churn) while still giving 4 independent WMMA accumulator chains (16 WMMAs) in flight.

typedef __attribute__((ext_vector_type(2))) float v2f;
typedef __attribute__((ext_vector_type(4))) float v4f;
typedef __attribute__((ext_vector_type(8))) float v8f;

#define FF_N       4096
#define FF_WAVES   2
#define FF_THREADS (FF_WAVES * 32)

__device__ __forceinline__ float hsign(int a, int b) {
  // H16[a][b] = (-1)^popcount(a & b)  (natural-order Walsh-Hadamard)
  return (__popc(a & b) & 1) ? -1.0f : 1.0f;
}

struct HFrag {
  v2f hf[4];  // 8 VGPRs, live for the whole kernel
};

// In-place unnormalized FWHT of buf[0..4095] (one row), executed by one wave.
// Row viewed as cube X[a][b][c], idx = a*256 + b*16 + c; apply H16 along each axis.
__device__ __forceinline__ void fwht4096_wave(float* buf, int lane, const HFrag& H) {
  const int hi = lane >> 4;  // lane half: K-range selector for A/B fragments
  const int q  = lane & 15;  // M (for A) or N (for B/C/D)

  // ---- pass 0: contiguous axis c:  T <- T @ H   (A = data tile, B = H) ----
#pragma unroll 4
  for (int t = 0; t < 16; ++t) {
    float* tp = buf + t * 256;
    v8f acc = {};
#pragma unroll
    for (int c = 0; c < 4; ++c) {
      const int ks = 4 * c + 2 * hi;
      v2f a = *(const v2f*)(tp + q * 16 + ks);  // ds_load_b64 (K-pair contiguous)
      acc = __builtin_amdgcn_wmma_f32_16x16x4_f32(false, a, false, H.hf[c],
                                                  (short)0, acc, false, false);
    }
#pragma unroll
    for (int v = 0; v < 8; ++v) tp[(v + 8 * hi) * 16 + q] = acc[v];
  }

  // ---- pass 1: stride-16 axis b:  T <- H @ T   (A = H, B = data tile) ----
#pragma unroll 4
  for (int t = 0; t < 16; ++t) {
    float* tp = buf + t * 256;
    v8f acc = {};
#pragma unroll
    for (int c = 0; c < 4; ++c) {
      const int ks = 4 * c + 2 * hi;
      v2f b;
      b.x = tp[ks * 16 + q];
      b.y = tp[(ks + 1) * 16 + q];
      acc = __builtin_amdgcn_wmma_f32_16x16x4_f32(false, H.hf[c], false, b,
                                                  (short)0, acc, false, false);
    }
#pragma unroll
    for (int v = 0; v < 8; ++v) tp[(v + 8 * hi) * 16 + q] = acc[v];
  }

  // ---- pass 2: stride-256 axis a:  T <- H @ T ----
#pragma unroll 4
  for (int t = 0; t < 16; ++t) {
    float* tp = buf + t * 16;
    v8f acc = {};
#pragma unroll
    for (int c = 0; c < 4; ++c) {
      const int ks = 4 * c + 2 * hi;
      v2f b;
      b.x = tp[ks * 256 + q];
      b.y = tp[(ks + 1) * 256 + q];
      acc = __builtin_amdgcn_wmma_f32_16x16x4_f32(false, H.hf[c], false, b,
                                                  (short)0, acc, false, false);
    }
#pragma unroll
    for (int v = 0; v < 8; ++v) tp[(v + 8 * hi) * 256 + q] = acc[v];
  }
}

__launch_bounds__(FF_THREADS)
__global__ void fastfood_kernel(const float* __restrict__ x,
                                const float* __restrict__ diag_s,
                                const float* __restrict__ diag_g,
                                const float* __restrict__ diag_b,
                                const int*   __restrict__ perm,
                                float* __restrict__ out) {
  __shared__ float lds[FF_WAVES * 2 * FF_N];  // 64 KB: two 16KB buffers per wave
  const int wave = threadIdx.x >> 5;
  const int lane = threadIdx.x & 31;
  const size_t row = (size_t)blockIdx.x * FF_WAVES + wave;

  float* bufA = lds + wave * (2 * FF_N);
  float* bufB = bufA + FF_N;
  const float* xr   = x + row * (size_t)FF_N;
  float*       outr = out + row * (size_t)FF_N;

  // H16 fragments for the four K-chunks (K = 4c + {2*hi, 2*hi+1}); computed once.
  // H is symmetric, so A-layout and B-layout fragments are identical.
  HFrag H;
  {
    const int hi = lane >> 4;
    const int q  = lane & 15;
#pragma unroll
    for (int c = 0; c < 4; ++c) {
      const int ks = 4 * c + 2 * hi;
      H.hf[c].x = hsign(q, ks);
      H.hf[c].y = hsign(q, ks + 1);
    }
  }

  // Stage row * diag_b into LDS (coalesced b128 loads, 512 B per wave per iter).
  for (int i = 0; i < FF_N / 128; ++i) {
    const int off = (i * 32 + lane) * 4;
    v4f xv = *(const v4f*)(xr + off);
    v4f bv = *(const v4f*)(diag_b + off);
    *(v4f*)(bufA + off) = xv * bv;
  }

  fwht4096_wave(bufA, lane, H);  // first FWHT (unnormalized)

  // Permute (gather) + diag_g into second buffer: y[k] = yA[perm[k]] * g[k]
  for (int i = 0; i < FF_N / 128; ++i) {
    const int off = (i * 32 + lane) * 4;
    int4 p = *(const int4*)(perm + off);
    v4f g  = *(const v4f*)(diag_g + off);
    v4f r;
    r.x = bufA[p.x] * g.x;
    r.y = bufA[p.y] * g.y;
    r.z = bufA[p.z] * g.z;
    r.w = bufA[p.w] * g.w;
    *(v4f*)(bufB + off) = r;
  }

  fwht4096_wave(bufB, lane, H);  // second FWHT (unnormalized)

  // Both 1/sqrt(4096) normalizations folded into one exact 2^-12 scale.
  const float sc = 1.0f / 4096.0f;
  for (int i = 0; i < FF_N / 128; ++i) {
    const int off = (i * 32 + lane) * 4;
    v4f yv = *(const v4f*)(bufB + off);
    v4f sv = *(const v4f*)(diag_s + off);
    *(v4f*)(outr + off) = yv * sv * sc;
  }
}

extern "C" void kernel_launch(void* const* d_in, const int* in_sizes, int n_in,
                              void* d_out, int out_size, void* d_ws, size_t ws_size,
                              hipStream_t stream) {
  // setup_inputs order: x, diag_s, diag_g, diag_b, perm_idx
  const float* x      = (const float*)d_in[0];
  const float* diag_s = (const float*)d_in[1];
  const float* diag_g = (const float*)d_in[2];
  const float* diag_b = (const float*)d_in[3];
  const int*   perm   = (const int*)d_in[4];
  float* out = (float*)d_out;

  const int rows = in_sizes[0] / FF_N;  // 16384
  dim3 grid(rows / FF_WAVES);
  dim3 block(FF_THREADS);
  fastfood_kernel<<<grid, block, 0, stream>>>(x, diag_s, diag_g, diag_b, perm, out);
}